// DiffMarkowitzLayer_14585708937862
// MI455X (gfx1250) — compile-verified
//
#include <hip/hip_runtime.h>
#include <math.h>

#define NDIM 256
#define BDIM 2048
#define NITERS 200
#define ROWS_PER_WG 64
#define BLOCK 128
#define WSTR 260          // ls_w/ls_p row stride in floats (bank-conflict padding)
#define TRSTR 20          // transpose scratch row stride (80B, 16B aligned)
#define POW_ITERS 24
#define MICHELOT_PASSES 8

typedef float v2f __attribute__((ext_vector_type(2)));
typedef float v8f __attribute__((ext_vector_type(8)));
typedef unsigned int v4u __attribute__((ext_vector_type(4)));
typedef int v4i __attribute__((ext_vector_type(4)));
typedef int v8i __attribute__((ext_vector_type(8)));

// Issue one TDM 2D tile load: Sigma[:, 16j : 16j+16] (256 rows x 64B) -> LDS.
// D# per CDNA5 ISA ch.8: group0 = {count/flags, lds_addr, global_addr lo, addr hi|type=2},
// group1 = {mask/data_size, tensor_dim0, dim0hi/tensor_dim1, dim1hi/tile_dim0,
//           tile_dim1/tile_dim2, dim0_stride lo, dim0_stride hi/dim1_stride lo, dim1_stride hi}.
__device__ static inline void tdm_load_sigma_tile(const float* Sigma, int j, float* lds_dst)
{
    unsigned long long gaddr = (unsigned long long)(uintptr_t)(Sigma + j * 16);
    unsigned int laddr = (unsigned int)(uintptr_t)lds_dst;   // addr[31:0] == LDS byte offset
    v4u g0;
    g0[0] = 1u;                                              // count=1, user mode, no gather
    g0[1] = laddr;                                           // lds_addr
    g0[2] = (unsigned int)gaddr;                             // global_addr[31:0]
    g0[3] = (unsigned int)((gaddr >> 32) & 0x1FFFFFFu)       // global_addr[56:32]
          | (2u << 30);                                      // type = 2 (image)
    v8i g1;
    g1[0] = (int)(2u << 16);        // workgroup_mask=0, data_size=2 (4B), no flags
    g1[1] = (int)(16u << 16);       // atomic_barrier_addr=0 ; tensor_dim0[15:0] = 16
    g1[2] = (int)(256u << 16);      // tensor_dim0[31:16]=0 ; tensor_dim1[15:0] = 256
    g1[3] = (int)(16u << 16);       // tensor_dim1[31:16]=0 ; tile_dim0 = 16
    g1[4] = 256;                    // tile_dim1 = 256 ; tile_dim2 = 0 (2D)
    g1[5] = 256;                    // tensor_dim0_stride[31:0] = 256 elements (row pitch)
    g1[6] = 0;                      // dim0_stride[47:32]=0 ; dim1_stride[15:0]=0
    g1[7] = 1;                      // dim1_stride[47:16]=1 -> 65536 elements (unused for 2D)
    v4i z4 = {0, 0, 0, 0};          // groups 2/3 unused (tile_dim2 = 0)
    v8i z8 = {0, 0, 0, 0, 0, 0, 0, 0};
    __builtin_amdgcn_tensor_load_to_lds(g0, g1, z4, z4, z8, 0);
}

__launch_bounds__(BLOCK)
__global__ void markowitz_fista_kernel(const float* __restrict__ p,
                                       const float* __restrict__ Sigma,
                                       float* __restrict__ out)
{
    __shared__ float ls_sig[2][NDIM * 16];       // 32KB double-buffered Sigma col-tile
    __shared__ float ls_ypk[4 * NDIM * 16];      // 64KB: per-wave packed y (WMMA A layout)
    __shared__ float ls_w[ROWS_PER_WG * WSTR];   // ~65KB: w iterate, row-major padded
    __shared__ float ls_p[ROWS_PER_WG * WSTR];   // ~65KB: p cached for 200 iterations
    __shared__ float ls_tr[4 * 16 * TRSTR];      // per-wave 16x16 transpose scratch (padded)
    __shared__ float ls_z[NDIM];
    __shared__ float ls_zn[NDIM];
    __shared__ float ls_red[BLOCK];

    const int tid  = threadIdx.x;
    const int wv   = tid >> 5;     // wave id 0..3
    const int ln   = tid & 31;     // lane
    const int m    = ln & 15;      // row (A) / col (B,C) index within tile
    const int h    = ln >> 4;      // lane half
    const int wg   = blockIdx.x;
    const int row0 = wg * ROWS_PER_WG + wv * 16;   // wave's first global batch row
    (void)row0;

    // Kick off the first Sigma tile DMA immediately; it overlaps the whole
    // power-iteration + init phase below.
    if (wv == 0) tdm_load_sigma_tile(Sigma, 0, &ls_sig[0][0]);

    // stage p for this WG's rows into LDS (reused 200x)
    for (int c = tid; c < ROWS_PER_WG * (NDIM / 4); c += BLOCK) {
        int r = c >> 6, g = c & 63;
        float4 d = *(const float4*)(p + (wg * ROWS_PER_WG + r) * NDIM + g * 4);
        *(float4*)(ls_p + r * WSTR + g * 4) = d;
    }

    //================ power iteration: L = lambda_max(Sigma) ================
    for (int i = tid; i < NDIM; i += BLOCK) ls_z[i] = 1.0f;
    __syncthreads();

    const float* rp0 = Sigma + (2 * tid) * NDIM;
    const float* rp1 = rp0 + NDIM;
    for (int it = 0; it < POW_ITERS; ++it) {
        float a0 = 0.0f, a1 = 0.0f;
        for (int c = 0; c < NDIM; c += 4) {
            float4 s0 = *(const float4*)(rp0 + c);
            float4 s1 = *(const float4*)(rp1 + c);
            float4 zz = *(const float4*)(ls_z + c);
            a0 += s0.x * zz.x + s0.y * zz.y + s0.z * zz.z + s0.w * zz.w;
            a1 += s1.x * zz.x + s1.y * zz.y + s1.z * zz.z + s1.w * zz.w;
        }
        ls_red[tid] = a0 * a0 + a1 * a1;
        for (int s = BLOCK / 2; s > 0; s >>= 1) {
            __syncthreads();
            if (tid < s) ls_red[tid] += ls_red[tid + s];
        }
        __syncthreads();
        float inv = 1.0f / sqrtf(ls_red[0]);
        __syncthreads();
        ls_z[2 * tid]     = a0 * inv;
        ls_z[2 * tid + 1] = a1 * inv;
        __syncthreads();
    }
    // Rayleigh quotient: lam = z'Sz / z'z
    {
        float a0 = 0.0f, a1 = 0.0f;
        for (int c = 0; c < NDIM; c += 4) {
            float4 s0 = *(const float4*)(rp0 + c);
            float4 s1 = *(const float4*)(rp1 + c);
            float4 zz = *(const float4*)(ls_z + c);
            a0 += s0.x * zz.x + s0.y * zz.y + s0.z * zz.z + s0.w * zz.w;
            a1 += s1.x * zz.x + s1.y * zz.y + s1.z * zz.z + s1.w * zz.w;
        }
        float z0 = ls_z[2 * tid], z1 = ls_z[2 * tid + 1];
        ls_red[tid] = a0 * z0 + a1 * z1;               // numerator partial
        for (int s = BLOCK / 2; s > 0; s >>= 1) {
            __syncthreads();
            if (tid < s) ls_red[tid] += ls_red[tid + s];
        }
        __syncthreads();
        ls_zn[0] = ls_red[0];                          // stash numerator
        __syncthreads();
        ls_red[tid] = z0 * z0 + z1 * z1;               // denominator partial
        for (int s = BLOCK / 2; s > 0; s >>= 1) {
            __syncthreads();
            if (tid < s) ls_red[tid] += ls_red[tid + s];
        }
        __syncthreads();
    }
    const float lam = ls_zn[0] / ls_red[0];
    const float lr  = 1.0f / (lam * 1.0005f);          // safety margin vs power-iter underestimate
    __syncthreads();

    //================ init y = w = 1/N, t = 1 ================
    const float w0 = 1.0f / (float)NDIM;
    for (int i = tid; i < ROWS_PER_WG * WSTR; i += BLOCK) ls_w[i] = w0;
    float* ypk = ls_ypk + wv * (NDIM * 16);
    for (int i = ln; i < NDIM * 16; i += 32) ypk[i] = w0;
    float t = 1.0f;

    float vv[128];   // per-lane: row (ln&15), cols [h*128, h*128+128)

    // tile 0 must be resident before the first stage
    if (wv == 0) __builtin_amdgcn_s_wait_tensorcnt(0);
    __syncthreads();

    //================ FISTA main loop ================
    for (int it = 0; it < NITERS; ++it) {
        #pragma unroll
        for (int j = 0; j < 16; ++j) {
            const float* sig = &ls_sig[j & 1][0];
            // Prefetch next tile via TDM into the other buffer; overlaps the
            // 64 WMMAs below (j==15 fetches tile 0 of the next iteration, waited on before projection).
            if (wv == 0 && (it < NITERS - 1 || j < 15))
                tdm_load_sigma_tile(Sigma, (j + 1) & 15, &ls_sig[(j + 1) & 1][0]);

            // ---- C init: p tile in C layout (from LDS) ----
            v8f acc;
            #pragma unroll
            for (int r = 0; r < 8; ++r)
                acc[r] = ls_p[(wv * 16 + r + 8 * h) * WSTR + j * 16 + m];

            // ---- K loop: 64 x V_WMMA_F32_16X16X4_F32 ----
            #pragma unroll 4
            for (int k = 0; k < NDIM; k += 4) {
                v2f a = *(const v2f*)(ypk + k * 16 + h * 32 + m * 2);   // A: y[m][k+2h .. +1]
                v2f b;
                b.x = sig[(k + 2 * h) * 16 + m];                        // B: S[k+2h][16j+m]
                b.y = sig[(k + 2 * h + 1) * 16 + m];                    // B: S[k+2h+1][16j+m]
                acc = __builtin_amdgcn_wmma_f32_16x16x4_f32(
                        false, a, false, b, (short)0, acc, false, false);
            }

            // ---- v = y - lr*grad (C layout), transpose to row layout via LDS ----
            float* tr = ls_tr + wv * 16 * TRSTR;
            #pragma unroll
            for (int r = 0; r < 8; ++r) {
                int col = j * 16 + m;
                float yv = ypk[(col >> 1) * 32 + (r + 8 * h) * 2 + (col & 1)];
                tr[(r + 8 * h) * TRSTR + m] = yv - lr * acc[r];
            }
            __asm__ volatile("" ::: "memory");     // keep DS store->load order (DS is in-order per wave)
            if ((j >> 3) == h) {
                int jl = j & 7;
                #pragma unroll
                for (int q = 0; q < 16; ++q)
                    vv[jl * 16 + q] = tr[m * TRSTR + q];
            }

            // next tile ready? (wave 0 tracks the DMA) then publish to all waves
            if (wv == 0) __builtin_amdgcn_s_wait_tensorcnt(0);
            __syncthreads();
        }

        // ---- simplex projection per row: Michelot fixed point on theta ----
        float s = 0.0f;
        #pragma unroll
        for (int i = 0; i < 128; ++i) s += vv[i];
        s += __shfl_xor(s, 16, 32);
        float theta = (s - 1.0f) * (1.0f / (float)NDIM);
        #pragma unroll 1
        for (int pass = 0; pass < MICHELOT_PASSES; ++pass) {
            float ss = 0.0f, cc = 0.0f;
            #pragma unroll
            for (int i = 0; i < 128; ++i) {
                bool gt = vv[i] > theta;
                ss += gt ? vv[i] : 0.0f;
                cc += gt ? 1.0f : 0.0f;
            }
            ss += __shfl_xor(ss, 16, 32);
            cc += __shfl_xor(cc, 16, 32);
            theta = (ss - 1.0f) / fmaxf(cc, 1.0f);
        }

        // ---- FISTA momentum + write-back (w row-major, y packed A-layout) ----
        float tn = 0.5f * (1.0f + sqrtf(1.0f + 4.0f * t * t));
        float cm = (t - 1.0f) / tn;
        t = tn;
        float* wrow = ls_w + (wv * 16 + m) * WSTR + h * 128;
        #pragma unroll
        for (int i = 0; i < 128; ++i) {
            float wn = fmaxf(vv[i] - theta, 0.0f);
            float wp = wrow[i];
            float yn = wn + cm * (wn - wp);
            wrow[i] = wn;
            int col = h * 128 + i;
            ypk[(col >> 1) * 32 + m * 2 + (col & 1)] = yn;
        }
        // stage j=15's trailing barrier fences the workgroup before Sigma reuse;
        // ypk/ls_w are wave-private so program order suffices here.
    }

    //================ write result ================
    __syncthreads();
    for (int c = tid; c < ROWS_PER_WG * (NDIM / 4); c += BLOCK) {
        int r = c >> 6, g = c & 63;
        float4 d = *(const float4*)(ls_w + r * WSTR + g * 4);
        *(float4*)(out + (wg * ROWS_PER_WG + r) * NDIM + g * 4) = d;
    }
}

extern "C" void kernel_launch(void* const* d_in, const int* in_sizes, int n_in,
                              void* d_out, int out_size, void* d_ws, size_t ws_size,
                              hipStream_t stream) {
    (void)in_sizes; (void)n_in; (void)d_ws; (void)ws_size; (void)out_size;
    const float* p_batch = (const float*)d_in[0];   // [2048, 256] f32
    const float* Sigma   = (const float*)d_in[1];   // [256, 256]  f32
    float* out           = (float*)d_out;           // [2048, 256] f32
    dim3 grid(BDIM / ROWS_PER_WG);                  // 32 workgroups
    dim3 block(BLOCK);                              // 128 threads = 4 waves
    markowitz_fista_kernel<<<grid, block, 0, stream>>>(p_batch, Sigma, out);
}